// BranchingGNN_57801669869677
// MI455X (gfx1250) — compile-verified
//
#include <hip/hip_runtime.h>

#define HDIM 64
#define ROUNDS 3

typedef __attribute__((ext_vector_type(2))) float v2f;
typedef __attribute__((ext_vector_type(8))) float v8f;

// ---------------------------------------------------------------------------
// out[M x 64] = act( A[M x K] @ B[K x 64] + bias  (+ resid) ), act = relu
// One 16-row tile per block. 128 threads = 4 waves; wave w computes the
// 16x16 output tile covering columns [16w, 16w+16) using V_WMMA_F32_16X16X4_F32.
// B (<= 64x64 f32 = 16 KB) is staged in LDS.
// ---------------------------------------------------------------------------
__global__ __launch_bounds__(128)
void gemm_bias_act_kernel(const float* __restrict__ A,
                          const float* __restrict__ B,
                          const float* __restrict__ bias,
                          const float* __restrict__ resid,   // may be null
                          float* __restrict__ out,
                          int M, int K) {
    __shared__ float ldsB[64 * HDIM];

    // Stage B into LDS (vectorized; K*HDIM is a multiple of 4)
    {
        const int n4 = (K * HDIM) >> 2;
        const float4* Bv = (const float4*)B;
        float4* Lv = (float4*)ldsB;
        for (int i = threadIdx.x; i < n4; i += blockDim.x) Lv[i] = Bv[i];
    }
    __syncthreads();

    const int tid  = threadIdx.x;
    const int wave = tid >> 5;
    const int lane = tid & 31;
    const int half = lane >> 4;   // 0: lanes 0-15, 1: lanes 16-31
    const int l16  = lane & 15;
    const int row0 = blockIdx.x * 16;
    const int nb   = wave * 16;   // column tile base

    // A row this lane feeds (D row m comes only from A row m, so clamping the
    // tail is safe: garbage only lands in rows we never store).
    int m = row0 + l16;
    if (m > M - 1) m = M - 1;
    const float* Arow = A + (size_t)m * K;

    v8f acc = {0.f, 0.f, 0.f, 0.f, 0.f, 0.f, 0.f, 0.f};

    for (int kb = 0; kb < K; kb += 4) {
        const int ka = kb + 2 * half;           // even -> 8B aligned
        const v2f a = *(const v2f*)(Arow + ka); // A[m][ka], A[m][ka+1]
        v2f b;
        b.x = ldsB[(ka + 0) * HDIM + nb + l16]; // B[ka  ][col]
        b.y = ldsB[(ka + 1) * HDIM + nb + l16]; // B[ka+1][col]
        acc = __builtin_amdgcn_wmma_f32_16x16x4_f32(
            /*neg_a=*/false, a, /*neg_b=*/false, b,
            /*c_mod=*/(short)0, acc, /*reuse_a=*/false, /*reuse_b=*/false);
    }

    const int col = nb + l16;
    const float bv = bias[col];
#pragma unroll
    for (int r = 0; r < 8; ++r) {
        const int rowg = row0 + r + 8 * half;   // C/D layout: VGPR r -> M=r (+8 for hi half)
        if (rowg < M) {
            float v = acc[r] + bv;
            if (resid) v += resid[(size_t)rowg * HDIM + col];
            out[(size_t)rowg * HDIM + col] = fmaxf(v, 0.f);
        }
    }
}

// ---------------------------------------------------------------------------
// msg[dst_idx[e]][:] += h_src[src_idx[e]][:]   (segment_sum via L2 f32 atomics)
// One wave per edge; each lane handles 2 consecutive columns (b64 gather,
// 2x global_atomic_add_f32). Arrays live in L2 (h buffers << 192 MB).
// ---------------------------------------------------------------------------
__global__ __launch_bounds__(256)
void edge_scatter_kernel(const float* __restrict__ h_src,
                         const int* __restrict__ src_idx,
                         const int* __restrict__ dst_idx,
                         float* __restrict__ msg,
                         int E) {
    const int wavesPerBlock = blockDim.x >> 5;
    const int e    = blockIdx.x * wavesPerBlock + (threadIdx.x >> 5);
    const int lane = threadIdx.x & 31;
    if (e >= E) return;

    // Prefetch the index stream ahead (gfx1250 global_prefetch_b8).
    __builtin_prefetch(src_idx + e + 4096, 0, 0);
    __builtin_prefetch(dst_idx + e + 4096, 0, 0);

    const int s = src_idx[e];
    const int d = dst_idx[e];
    const v2f val = *(const v2f*)(h_src + (size_t)s * HDIM + 2 * lane);
    float* dp = msg + (size_t)d * HDIM + 2 * lane;
    __hip_atomic_fetch_add(dp,     val.x, __ATOMIC_RELAXED, __HIP_MEMORY_SCOPE_AGENT);
    __hip_atomic_fetch_add(dp + 1, val.y, __ATOMIC_RELAXED, __HIP_MEMORY_SCOPE_AGENT);
}

// ---------------------------------------------------------------------------
__global__ void zero4_kernel(float4* __restrict__ p, int n4) {
    const int i = blockIdx.x * blockDim.x + threadIdx.x;
    if (i < n4) p[i] = make_float4(0.f, 0.f, 0.f, 0.f);
}

// scores[v] = h_var[v] . W_score + b_score
__global__ void score_kernel(const float* __restrict__ h_var,
                             const float* __restrict__ W_score,
                             const float* __restrict__ b_score,
                             float* __restrict__ out, int V) {
    const int v = blockIdx.x * blockDim.x + threadIdx.x;
    if (v >= V) return;
    const float4* hp = (const float4*)(h_var + (size_t)v * HDIM);
    const float4* wp = (const float4*)W_score;
    float s = 0.f;
#pragma unroll
    for (int i = 0; i < HDIM / 4; ++i) {
        const float4 h = hp[i];
        const float4 w = wp[i];
        s += h.x * w.x + h.y * w.y + h.z * w.z + h.w * w.w;
    }
    out[v] = s + b_score[0];
}

// ---------------------------------------------------------------------------
extern "C" void kernel_launch(void* const* d_in, const int* in_sizes, int n_in,
                              void* d_out, int out_size, void* d_ws, size_t ws_size,
                              hipStream_t stream) {
    const float* var_feat = (const float*)d_in[0];
    const float* con_feat = (const float*)d_in[1];
    const float* W_var    = (const float*)d_in[2];
    const float* b_var    = (const float*)d_in[3];
    const float* W_con    = (const float*)d_in[4];
    const float* b_con    = (const float*)d_in[5];
    const float* W_v2c    = (const float*)d_in[6];
    const float* b_v2c    = (const float*)d_in[7];
    const float* W_c2v    = (const float*)d_in[8];
    const float* b_c2v    = (const float*)d_in[9];
    const float* W_score  = (const float*)d_in[10];
    const float* b_score  = (const float*)d_in[11];
    const int*   var_idx  = (const int*)d_in[12];
    const int*   con_idx  = (const int*)d_in[13];

    const int V = in_sizes[0] / 32;   // var_feat is [V, 32]
    const int C = in_sizes[1] / 32;   // constr_feat is [C, 32]
    const int E = in_sizes[12];

    // Workspace layout: h_var [V*64] | h_con [C*64] | msg [V*64] (V >= C)
    float* h_var = (float*)d_ws;
    float* h_con = h_var + (size_t)V * HDIM;
    float* msg   = h_con + (size_t)C * HDIM;

    const dim3 gblk(128);
    const int vTiles = (V + 15) / 16;
    const int cTiles = (C + 15) / 16;
    const int eBlocks = (E + 7) / 8;              // 8 waves (edges) per 256-thr block
    const int zC = (C * HDIM) >> 2;
    const int zV = (V * HDIM) >> 2;

    // Embeddings
    gemm_bias_act_kernel<<<vTiles, gblk, 0, stream>>>(var_feat, W_var, b_var, nullptr, h_var, V, 32);
    gemm_bias_act_kernel<<<cTiles, gblk, 0, stream>>>(con_feat, W_con, b_con, nullptr, h_con, C, 32);

    for (int r = 0; r < ROUNDS; ++r) {
        // var -> constr
        zero4_kernel<<<(zC + 255) / 256, 256, 0, stream>>>((float4*)msg, zC);
        edge_scatter_kernel<<<eBlocks, 256, 0, stream>>>(h_var, var_idx, con_idx, msg, E);
        gemm_bias_act_kernel<<<cTiles, gblk, 0, stream>>>(msg, W_v2c, b_v2c, h_con, h_con, C, HDIM);
        // constr -> var
        zero4_kernel<<<(zV + 255) / 256, 256, 0, stream>>>((float4*)msg, zV);
        edge_scatter_kernel<<<eBlocks, 256, 0, stream>>>(h_con, con_idx, var_idx, msg, E);
        gemm_bias_act_kernel<<<vTiles, gblk, 0, stream>>>(msg, W_c2v, b_c2v, h_var, h_var, V, HDIM);
    }

    score_kernel<<<(V + 255) / 256, 256, 0, stream>>>(h_var, W_score, b_score, (float*)d_out, V);
}